// ResGraphConvPool_73890617360438
// MI455X (gfx1250) — compile-verified
//
#include <hip/hip_runtime.h>

typedef __attribute__((ext_vector_type(2))) float v2f;
typedef __attribute__((ext_vector_type(8))) float v8f;

#define BATCH   8
#define NPTS    4096
#define CH      64
#define KNN     8
#define NLAYERS 12
#define BN_EPSF 1e-5f
#define SLOPE   0.01f

// ---------------------------------------------------------------------------
// Kernel 1: brute-force KNN (self included), per batch. One thread per query,
// candidate xyz tiles staged through LDS, 8-best kept in registers.
// ---------------------------------------------------------------------------
__global__ __launch_bounds__(128) void knn_kernel(const float* __restrict__ xyz,
                                                  int* __restrict__ idx_out)
{
    const int b = blockIdx.y;
    const int q = blockIdx.x * 128 + (int)threadIdx.x;
    const float* xb = xyz + (size_t)b * 3 * NPTS;
    const float qx = xb[q], qy = xb[NPTS + q], qz = xb[2 * NPTS + q];

    float bd[KNN];
    int   bi[KNN];
#pragma unroll
    for (int s = 0; s < KNN; ++s) { bd[s] = 3.4e38f; bi[s] = 0; }

    __shared__ float sx[256], sy[256], sz[256];
    for (int tile = 0; tile < NPTS; tile += 256) {
        for (int t = threadIdx.x; t < 256; t += 128) {
            sx[t] = xb[tile + t];
            sy[t] = xb[NPTS + tile + t];
            sz[t] = xb[2 * NPTS + tile + t];
        }
        __syncthreads();
        for (int j = 0; j < 256; ++j) {
            const float dx = qx - sx[j], dy = qy - sy[j], dz = qz - sz[j];
            const float d2 = dx * dx + dy * dy + dz * dz;
            if (d2 < bd[KNN - 1]) {           // rare path: insert into sorted 8-best
                float d = d2; int ci = tile + j;
#pragma unroll
                for (int s = 0; s < KNN; ++s) {
                    if (d < bd[s]) {
                        float td = bd[s]; int ti = bi[s];
                        bd[s] = d; bi[s] = ci; d = td; ci = ti;
                    }
                }
            }
        }
        __syncthreads();
    }
    int* o = idx_out + ((size_t)b * NPTS + q) * KNN;
#pragma unroll
    for (int s = 0; s < KNN; ++s) o[s] = bi[s];
}

// ---------------------------------------------------------------------------
// Kernel 2: BN batch statistics per channel -> fused (scale, shift) so that
// h = p*scale + shift. One block per channel, reduction over b*n = 32768.
// ---------------------------------------------------------------------------
__global__ __launch_bounds__(256) void bn_stats_kernel(const float* __restrict__ p,
                                                       const float* __restrict__ gamma,
                                                       const float* __restrict__ beta,
                                                       float* __restrict__ stats)
{
    const int c = blockIdx.x;
    __shared__ float ssum[256], ssq[256];
    float s = 0.f, s2 = 0.f;
    for (int e = threadIdx.x; e < BATCH * NPTS; e += 256) {
        const int b = e >> 12, i = e & (NPTS - 1);
        const float v = p[((size_t)b * CH + c) * NPTS + i];
        s += v; s2 += v * v;
    }
    ssum[threadIdx.x] = s; ssq[threadIdx.x] = s2;
    __syncthreads();
    for (int st = 128; st > 0; st >>= 1) {
        if ((int)threadIdx.x < st) {
            ssum[threadIdx.x] += ssum[threadIdx.x + st];
            ssq[threadIdx.x]  += ssq[threadIdx.x + st];
        }
        __syncthreads();
    }
    if (threadIdx.x == 0) {
        const float inv  = 1.0f / (float)(BATCH * NPTS);
        const float mean = ssum[0] * inv;
        const float var  = ssq[0] * inv - mean * mean;
        const float sc   = gamma[c] * rsqrtf(var + BN_EPSF);
        stats[c]      = sc;
        stats[CH + c] = beta[c] - mean * sc;   // shift
    }
}

// ---------------------------------------------------------------------------
// Kernel 3: m[c,j] = ( f(p[c,j]) + sum_k f(p[c,idx[j,k]]) ) / 9,
// f = LeakyReLU(BN). One thread per point, loops over channels so the 8
// neighbor indices are loaded once per thread.
// ---------------------------------------------------------------------------
__device__ __forceinline__ float bn_lrelu(float x, float sc, float sh)
{
    const float h = fmaf(x, sc, sh);
    return h >= 0.f ? h : SLOPE * h;
}

__global__ __launch_bounds__(256) void gather_avg_kernel(const float* __restrict__ p,
                                                         const int* __restrict__ idx,
                                                         const float* __restrict__ stats,
                                                         float* __restrict__ m)
{
    const int b = blockIdx.y;
    const int j = blockIdx.x * 256 + (int)threadIdx.x;

    __shared__ float sstats[2 * CH];
    if (threadIdx.x < 2 * CH) sstats[threadIdx.x] = stats[threadIdx.x];
    __syncthreads();

    int nb[KNN];
    const int* ir = idx + ((size_t)b * NPTS + j) * KNN;
#pragma unroll
    for (int k = 0; k < KNN; ++k) nb[k] = ir[k];

    const float* pb = p + (size_t)b * CH * NPTS;
    float*       mb = m + (size_t)b * CH * NPTS;
    const float invk = 1.0f / (float)(KNN + 1);

    for (int c = 0; c < CH; ++c) {
        const float sc = sstats[c], sh = sstats[CH + c];
        const float* row = pb + (size_t)c * NPTS;
        float acc = bn_lrelu(row[j], sc, sh);
#pragma unroll
        for (int k = 0; k < KNN; ++k) acc += bn_lrelu(row[nb[k]], sc, sh);
        mb[(size_t)c * NPTS + j] = acc * invk;
    }
}

// ---------------------------------------------------------------------------
// Kernel 4: out = W(64x64) @ m(64 x NPTS) + bias + shortcut, via
// V_WMMA_F32_16X16X4_F32. 512 threads = 16 waves; each wave owns one 16x16
// output tile of the block's 64(M) x 64(N) macro-tile; 16 k-steps of K=4.
//
// Layouts (CDNA5 ISA):
//   A 16x4:  lane m = row (lanes 0-15 and 16-31 both M=0..15);
//            v2f = K { 2*hi, 2*hi+1 } where hi = lane>>4.
//   B 4x16:  col = lane&15; v2f = rows K { 2*hi, 2*hi+1 }.
//   C/D:     acc[v] holds M = tm + v + 8*hi, N = n0 + (lane&15).
// ---------------------------------------------------------------------------
__global__ __launch_bounds__(512) void gemm_wmma_kernel(const float* __restrict__ m,
                                                        const float* __restrict__ W,
                                                        const float* __restrict__ bias,
                                                        const float* __restrict__ shortcut,
                                                        float* __restrict__ out)
{
    const int lane = threadIdx.x & 31;
    const int wv   = threadIdx.x >> 5;          // 0..15
    const int tm   = (wv >> 2) * 16;            // 0,16,32,48  (output-channel tile)
    const int tn   = (wv & 3) * 16;             // 0,16,32,48  (point tile within 64)
    const int b    = blockIdx.y;
    const int n0   = blockIdx.x * 64 + tn;
    const int hi   = lane >> 4;
    const int lo   = lane & 15;

    const float* mb = m + (size_t)b * CH * NPTS;
    const int arow  = tm + lo;

    v8f acc = {};
#pragma unroll
    for (int k = 0; k < CH; k += 4) {
        const int ka = k + 2 * hi;
        v2f a, bv;
        a.x  = W[(size_t)arow * CH + ka];
        a.y  = W[(size_t)arow * CH + ka + 1];
        bv.x = mb[(size_t)ka * NPTS + n0 + lo];
        bv.y = mb[(size_t)(ka + 1) * NPTS + n0 + lo];
        acc = __builtin_amdgcn_wmma_f32_16x16x4_f32(
            /*neg_a=*/false, a, /*neg_b=*/false, bv,
            /*c_mod=*/(short)0, acc, /*reuse_a=*/false, /*reuse_b=*/false);
    }

    const int col   = n0 + lo;
    const float* sb = shortcut + (size_t)b * CH * NPTS;
    float*       ob = out      + (size_t)b * CH * NPTS;
#pragma unroll
    for (int v = 0; v < 8; ++v) {
        const int o = tm + v + 8 * hi;
        ob[(size_t)o * NPTS + col] = acc[v] + bias[o] + sb[(size_t)o * NPTS + col];
    }
}

// ---------------------------------------------------------------------------
// Host-side launcher. Workspace layout:
//   [0,       1 MiB)  idx   : int  [8][4096][8]
//   [1 MiB,   9 MiB)  m     : f32  [8][64][4096]
//   [9 MiB,  17 MiB)  p ping: f32  [8][64][4096]
//   [17 MiB, +512 B)  stats : f32  [2][64]
// Block outputs alternate ws_p / d_out so block 11 lands in d_out.
// ---------------------------------------------------------------------------
extern "C" void kernel_launch(void* const* d_in, const int* in_sizes, int n_in,
                              void* d_out, int out_size, void* d_ws, size_t ws_size,
                              hipStream_t stream)
{
    (void)in_sizes; (void)n_in; (void)out_size; (void)ws_size;

    const float* xyz    = (const float*)d_in[0];
    const float* points = (const float*)d_in[1];
    const float* conv_w = (const float*)d_in[2];
    const float* conv_b = (const float*)d_in[3];
    const float* gamma  = (const float*)d_in[4];
    const float* beta   = (const float*)d_in[5];
    float* out = (float*)d_out;

    char* ws = (char*)d_ws;
    int*   ws_idx   = (int*)ws;
    float* ws_m     = (float*)(ws + (size_t)(1u << 20));
    float* ws_p     = (float*)(ws + (size_t)(9u << 20));
    float* ws_stats = (float*)(ws + (size_t)(17u << 20));

    // fixed neighbor graph, computed once
    knn_kernel<<<dim3(NPTS / 128, BATCH), 128, 0, stream>>>(xyz, ws_idx);

    for (int t = 0; t < NLAYERS; ++t) {
        const float* pin  = (t == 0) ? points : ((t & 1) ? ws_p : out);
        float*       pout = (t & 1) ? out : ws_p;

        bn_stats_kernel<<<CH, 256, 0, stream>>>(pin, gamma + t * CH, beta + t * CH, ws_stats);
        gather_avg_kernel<<<dim3(NPTS / 256, BATCH), 256, 0, stream>>>(pin, ws_idx, ws_stats, ws_m);
        gemm_wmma_kernel<<<dim3(NPTS / 64, BATCH), 512, 0, stream>>>(
            ws_m, conv_w + (size_t)t * CH * CH, conv_b + t * CH, pin, pout);
    }
}